// INFWhereMix_28802050687329
// MI455X (gfx1250) — compile-verified
//
#include <hip/hip_runtime.h>

typedef float v2f __attribute__((ext_vector_type(2)));
typedef float v8f __attribute__((ext_vector_type(8)));

#define NB   128   // batch
#define NW   8192  // spatial
#define NC   64    // in channels (K)
#define NO   64    // out channels (N)
#define NP   6
#define NH   32

// ---------------------------------------------------------------------------
// Hypernetwork: per-sample W[b, 64, 64] and biases[b, 64] into workspace.
// One block per sample; tiny FLOP count, pure VALU.
// ---------------------------------------------------------------------------
__global__ __launch_bounds__(256) void hyper_kernel(
    const float* __restrict__ params,   // [B, 6]
    const float* __restrict__ fc1_w,    // [32, 6]
    const float* __restrict__ fc1_b,    // [32]
    const float* __restrict__ fc2_w,    // [4096, 32]
    const float* __restrict__ fc2_b,    // [4096]
    const float* __restrict__ biasfc_w, // [64, 6]
    const float* __restrict__ biasfc_b, // [64]
    float* __restrict__ Wout,           // [B, 4096]  (o*64 + c)
    float* __restrict__ biasOut)        // [B, 64]
{
    const int b = blockIdx.x;
    const int t = threadIdx.x;
    __shared__ float p[NP];
    __shared__ float h[NH];

    if (t < NP) p[t] = params[b * NP + t];
    __syncthreads();

    if (t < NH) {
        float a = fc1_b[t];
        #pragma unroll
        for (int j = 0; j < NP; ++j) a = fmaf(p[j], fc1_w[t * NP + j], a);
        h[t] = fmaxf(a, 0.0f);
    }
    if (t < NO) {
        float a = biasfc_b[t];
        #pragma unroll
        for (int j = 0; j < NP; ++j) a = fmaf(p[j], biasfc_w[t * NP + j], a);
        biasOut[b * NO + t] = a;
    }
    __syncthreads();

    #pragma unroll
    for (int i = 0; i < 16; ++i) {
        const int oc = t + 256 * i;          // 0..4095
        float a = fc2_b[oc];
        #pragma unroll
        for (int j = 0; j < NH; ++j) a = fmaf(h[j], fc2_w[oc * NH + j], a);
        Wout[(size_t)b * (NO * NC) + oc] = a;
    }
}

// ---------------------------------------------------------------------------
// Main per-sample GEMM via V_WMMA_F32_16X16X4_F32.
// grid = (8, 128), block = 256 (8 wave32). Each wave: 8 tiles of 16 rows.
// Per wave: preload all B fragments (W matrix, invariant) once; per tile do
// 16 A-frag loads + 64 WMMA + 32 coalescing stores. Bias folded into C init.
// ---------------------------------------------------------------------------
__global__ __launch_bounds__(256) void conv_wmma_kernel(
    const float* __restrict__ x,       // [B, 8192, 64]
    const float* __restrict__ Wmat,    // [B, 64, 64]  (o-major)
    const float* __restrict__ biases,  // [B, 64]
    float* __restrict__ out)           // [B, 8192, 64]
{
    const int b        = blockIdx.y;
    const int waveId   = blockIdx.x * 8 + (threadIdx.x >> 5);  // 0..63 within sample
    const int lane     = threadIdx.x & 31;
    const int laneRow  = lane & 15;   // M (for A/C) or N (for B/C) selector
    const int laneHi   = lane >> 4;   // K-pair selector for A/B, M+8 for C/D

    const int tile0 = waveId * 8;     // 512 tiles of 16 rows per sample

    // ---- Preload B fragments: B[k=c][n16+laneRow] = Wmat[o][c] -------------
    const float* Wb = Wmat + (size_t)b * (NO * NC);
    v2f Bf[4][16];
    #pragma unroll
    for (int n = 0; n < 4; ++n) {
        const float* wrow = Wb + (size_t)(n * 16 + laneRow) * NC + 2 * laneHi;
        #pragma unroll
        for (int kk = 0; kk < 16; ++kk)
            Bf[n][kk] = *(const v2f*)(wrow + 4 * kk);
    }
    float biasv[4];
    #pragma unroll
    for (int n = 0; n < 4; ++n)
        biasv[n] = biases[b * NO + n * 16 + laneRow];

    const float* xb = x   + (size_t)b * NW * NC;
    float*       ob = out + (size_t)b * NW * NO;

    for (int t = 0; t < 8; ++t) {
        const int w0 = (tile0 + t) * 16;
        const float* xrow = xb + (size_t)(w0 + laneRow) * NC + 2 * laneHi;

        // Prefetch next tile's rows (global_prefetch_b8)
        if (t + 1 < 8) {
            __builtin_prefetch(xrow + 16 * NC, 0, 3);
            __builtin_prefetch(xrow + 16 * NC + 32, 0, 3);
        }

        // A fragments: A[M=laneRow][K = 4*kk + 2*laneHi + {0,1}]
        v2f A[16];
        #pragma unroll
        for (int kk = 0; kk < 16; ++kk)
            A[kk] = *(const v2f*)(xrow + 4 * kk);

        // C init = per-column bias (column is lane-indexed, same for all M)
        v8f acc[4];
        #pragma unroll
        for (int n = 0; n < 4; ++n) {
            #pragma unroll
            for (int r = 0; r < 8; ++r) acc[n][r] = biasv[n];
        }

        // 16 K-steps x 4 N-tiles; interleave N for 4 independent acc chains
        #pragma unroll
        for (int kk = 0; kk < 16; ++kk) {
            #pragma unroll
            for (int n = 0; n < 4; ++n) {
                acc[n] = __builtin_amdgcn_wmma_f32_16x16x4_f32(
                    false, A[kk], false, Bf[n][kk],
                    (short)0, acc[n], false, false);
            }
        }

        // Store: VGPR r holds out[w0 + r + 8*laneHi][n*16 + laneRow].
        // Each 16-lane half writes 64 contiguous bytes per (r, n).
        float* orow = ob + (size_t)(w0 + 8 * laneHi) * NO + laneRow;
        #pragma unroll
        for (int n = 0; n < 4; ++n) {
            #pragma unroll
            for (int r = 0; r < 8; ++r)
                orow[(size_t)r * NO + n * 16] = acc[n][r];
        }
    }
}

// ---------------------------------------------------------------------------
extern "C" void kernel_launch(void* const* d_in, const int* in_sizes, int n_in,
                              void* d_out, int out_size, void* d_ws, size_t ws_size,
                              hipStream_t stream) {
    const float* x        = (const float*)d_in[0];
    const float* params   = (const float*)d_in[1];
    const float* fc1_w    = (const float*)d_in[2];
    const float* fc1_b    = (const float*)d_in[3];
    const float* fc2_w    = (const float*)d_in[4];
    const float* fc2_b    = (const float*)d_in[5];
    const float* biasfc_w = (const float*)d_in[6];
    const float* biasfc_b = (const float*)d_in[7];
    float* out = (float*)d_out;

    float* Wout    = (float*)d_ws;              // 128*4096 floats = 2 MB
    float* biasOut = Wout + (size_t)NB * NO * NC;  // 128*64 floats

    hyper_kernel<<<NB, 256, 0, stream>>>(params, fc1_w, fc1_b, fc2_w, fc2_b,
                                         biasfc_w, biasfc_b, Wout, biasOut);

    dim3 grid(8, NB);
    conv_wmma_kernel<<<grid, 256, 0, stream>>>(x, Wout, biasOut, out);
}